// QNearestNeighbourManhattan_11819749998732
// MI455X (gfx1250) — compile-verified
//
#include <hip/hip_runtime.h>
#include <hip/hip_bf16.h>
#include <cstdint>

// Problem constants (from reference): B=4, V=2048, S=4, F=16, K=16
#define B_   4
#define V_   2048
#define S_   4
#define F_   16
#define K_   16
#define KM1_ 15

#define WAVES_PER_BLOCK 8
#define THREADS (WAVES_PER_BLOCK * 32)          // 256
#define BLOCKS_PER_BATCH (V_ / WAVES_PER_BLOCK) // 256
#define CAND_PER_LANE (V_ / 32)                 // 64

#define FLT_MAX_F 3.402823466e+38f
#define SENTINEL_KEY 0xFFFFFFFFFFFFFFFFull

// ---- CDNA5 async global->LDS copy (ASYNCcnt path, gfx1250) ----
__device__ __forceinline__ void async_load_b128_to_lds(const void* gaddr,
                                                       unsigned int ldsByteAddr) {
    // VDST = LDS byte address VGPR, VADDR = 64-bit global address pair, GV mode.
    asm volatile("global_load_async_to_lds_b128 %0, %1, off"
                 :: "v"(ldsByteAddr), "v"(gaddr)
                 : "memory");
}

__device__ __forceinline__ void wait_asynccnt0() {
#if __has_builtin(__builtin_amdgcn_s_wait_asynccnt)
    __builtin_amdgcn_s_wait_asynccnt(0);
#else
    asm volatile("s_wait_asynccnt 0x0" ::: "memory");
#endif
}

__global__ __launch_bounds__(THREADS)
void knn_manhattan_topk_kernel(const float* __restrict__ coords,   // (B,V,4)
                               const float* __restrict__ feats,    // (B,V,16)
                               const int*   __restrict__ active,   // (B,1)
                               float* __restrict__ outDist,        // (B,V,15)
                               float* __restrict__ outFeat)        // (B,V,15,16)
{
    __shared__ float4 sC[V_];                                          // 32 KB
    __shared__ unsigned long long sKeys[WAVES_PER_BLOCK][32 * K_];     // 32 KB

    const int tid  = threadIdx.x;
    const int lane = tid & 31;
    const int wave = tid >> 5;
    const int blk  = blockIdx.x;
    const int b    = blk / BLOCKS_PER_BATCH;
    const int i    = (blk % BLOCKS_PER_BATCH) * WAVES_PER_BLOCK + wave;

    const float* cb = coords + (size_t)b * V_ * S_;

    // ---- Stage this batch's coordinates (2048 x float4 = 32KB) into LDS ----
    #pragma unroll
    for (int it = 0; it < V_ / THREADS; ++it) {
        const int v = it * THREADS + tid;
        async_load_b128_to_lds(cb + (size_t)v * 4,
                               (unsigned int)(uintptr_t)(&sC[v]));
    }
    wait_asynccnt0();
    __syncthreads();

    const int  act       = active[b];
    const bool rowActive = (i < act);
    const float4 ci      = sC[i];

    // ---- Per-lane sorted top-16 keys (ascending). key = f32bits(d)<<32 | j ----
    unsigned long long key[K_];
    #pragma unroll
    for (int k = 0; k < K_; ++k) key[k] = SENTINEL_KEY;

    for (int t = 0; t < CAND_PER_LANE; ++t) {
        const int j = t * 32 + lane;              // lanes hit distinct j's
        const float4 cj = sC[j];
        float d = fabsf(ci.x - cj.x) + fabsf(ci.y - cj.y)
                + fabsf(ci.z - cj.z) + fabsf(ci.w - cj.w);
        if (!(rowActive && (j < act))) d = FLT_MAX_F;
        const unsigned long long cand =
            ((unsigned long long)__float_as_uint(d) << 32) | (unsigned int)j;
        if (cand < key[K_ - 1]) {
            bool ins = true;                      // branchless sorted insert
            #pragma unroll
            for (int k = K_ - 1; k >= 1; --k) {
                const bool shift = cand < key[k - 1];
                const unsigned long long nv = shift ? key[k - 1] : cand;
                key[k] = ins ? nv : key[k];
                ins = ins && shift;
            }
            if (ins) key[0] = cand;
        }
    }

    // ---- Dump per-lane lists to LDS, then 16-round wave tournament merge ----
    #pragma unroll
    for (int k = 0; k < K_; ++k) sKeys[wave][lane * K_ + k] = key[k];

    int p = 0;
    unsigned long long head = sKeys[wave][lane * K_];
    unsigned long long keep = SENTINEL_KEY;       // lane r keeps round-r winner
    for (int r = 0; r < K_; ++r) {
        unsigned long long m = head;
        #pragma unroll
        for (int s = 16; s >= 1; s >>= 1) {
            const unsigned long long o = __shfl_xor(m, s, 32);
            m = (o < m) ? o : m;
        }
        if (lane == r) keep = m;
        if (head == m) {                          // unique winner (idx embedded)
            ++p;
            head = (p < K_) ? sKeys[wave][lane * K_ + p] : SENTINEL_KEY;
        }
    }

    // ---- Emit: lanes 1..15 hold ranks 1..15 (rank 0 == self, dropped) ----
    const float od = __uint_as_float((unsigned int)(keep >> 32));
    const int   oi = (int)(unsigned int)(keep & 0xFFFFFFFFu);

    const size_t rowOut = ((size_t)b * V_ + i) * KM1_;
    if (lane >= 1 && lane < K_) outDist[rowOut + (lane - 1)] = od;

    const float* fb = feats + (size_t)b * V_ * F_;
    float*       of = outFeat + rowOut * F_;
    for (int k = 1; k < K_; ++k) {
        const int jj = __shfl(oi, k, 32);         // broadcast rank-k index
        if (lane < F_) of[(size_t)(k - 1) * F_ + lane] = fb[(size_t)jj * F_ + lane];
    }
}

extern "C" void kernel_launch(void* const* d_in, const int* in_sizes, int n_in,
                              void* d_out, int out_size, void* d_ws, size_t ws_size,
                              hipStream_t stream) {
    (void)in_sizes; (void)n_in; (void)out_size; (void)d_ws; (void)ws_size;
    const float* coords = (const float*)d_in[0];   // (B,V,S) f32
    const float* feats  = (const float*)d_in[1];   // (B,V,F) f32
    const int*   active = (const int*)d_in[2];     // (B,1)   i32

    float* outDist = (float*)d_out;                            // B*V*15
    float* outFeat = outDist + (size_t)B_ * V_ * KM1_;         // B*V*15*16

    dim3 grid(B_ * BLOCKS_PER_BATCH);
    dim3 block(THREADS);
    hipLaunchKernelGGL(knn_manhattan_topk_kernel, grid, block, 0, stream,
                       coords, feats, active, outDist, outFeat);
}